// L1Attn_4861902979525
// MI455X (gfx1250) — compile-verified
//
#include <hip/hip_runtime.h>
#include <math.h>

typedef float f4  __attribute__((ext_vector_type(4)));
typedef int   v4i __attribute__((ext_vector_type(4)));

#define NB 2
#define NS 2048
#define NH 8
#define ND 32
#define TQ 32
#define TT 32

#if defined(__has_builtin)
#if __has_builtin(__builtin_amdgcn_global_load_async_to_lds_b128)
#define ASYNC_BUILTIN 1
#endif
#endif

__device__ __forceinline__ void async_copy16(const float* gsrc, float* lds_dst) {
#if defined(ASYNC_BUILTIN)
  __builtin_amdgcn_global_load_async_to_lds_b128(
      (__attribute__((address_space(1))) v4i*)gsrc,
      (__attribute__((address_space(3))) v4i*)lds_dst,
      /*offset=*/0, /*cpol=*/0);
#else
  // inline-asm fallback: vdst = LDS byte address, vaddr = 64-bit global address
  unsigned lds_addr =
      (unsigned)(size_t)(__attribute__((address_space(3))) void*)lds_dst;
  asm volatile("global_load_async_to_lds_b128 %0, %1, off"
               :
               : "v"(lds_addr), "v"(gsrc)
               : "memory");
#endif
}

__device__ __forceinline__ void async_wait0() {
#if defined(__has_builtin)
#if __has_builtin(__builtin_amdgcn_s_wait_asynccnt)
  __builtin_amdgcn_s_wait_asynccnt(0);
#else
  asm volatile("s_wait_asynccnt 0" ::: "memory");
#endif
#else
  asm volatile("s_wait_asynccnt 0" ::: "memory");
#endif
}

__global__ __launch_bounds__(256) void l1_scores_kernel(
    const float* __restrict__ qg, const float* __restrict__ kg,
    float* __restrict__ out) {
  // q tile mirrors global layout: [qq][h][d]  (32*8*32 floats = 32 KB)
  __shared__ __align__(16) float lds_q[TQ * NH * ND];
  // output staging: [qq][t][h]               (32*32*8 floats = 32 KB)
  __shared__ __align__(16) float lds_o[TQ * TT * NH];

  const int tid  = threadIdx.x;
  const int h    = tid >> 5;   // wave id == head
  const int lane = tid & 31;   // lane   == key index within tile
  const int t0 = blockIdx.x * TT;
  const int q0 = blockIdx.y * TQ;
  const int b  = blockIdx.z;

  // ---- stage q tile into LDS via CDNA5 async DMA (8 x 16B per thread) ----
  const float* qsrc = qg + ((size_t)b * NS + q0) * (NH * ND);
#pragma unroll
  for (int pass = 0; pass < 8; ++pass) {
    const int idx4 = pass * 256 + tid;              // 2048 float4 total
    async_copy16(qsrc + idx4 * 4, &lds_q[idx4 * 4]);
  }

  // ---- while the async copy flies, pull this lane's k row into VGPRs ----
  const float* ksrc = kg + (((size_t)b * NS + (t0 + lane)) * NH + h) * ND;
  f4 kk[8];
#pragma unroll
  for (int j = 0; j < 8; ++j) kk[j] = ((const f4*)ksrc)[j];

  async_wait0();
  __syncthreads();

  // ---- compute: acc = sum_d |q - k|, 2 VALU per element (sub + add-with-|mod|)
  const float nscale = -0.17677669529663687f;  // -1/sqrt(32)
  for (int qq = 0; qq < TQ; ++qq) {
    const f4* qrow = (const f4*)&lds_q[(qq * NH + h) * ND];  // wave-uniform
    float a0 = 0.f, a1 = 0.f, a2 = 0.f, a3 = 0.f;
#pragma unroll
    for (int j = 0; j < 8; ++j) {
      const f4 qv = qrow[j];
      a0 += __builtin_fabsf(qv.x - kk[j].x);
      a1 += __builtin_fabsf(qv.y - kk[j].y);
      a2 += __builtin_fabsf(qv.z - kk[j].z);
      a3 += __builtin_fabsf(qv.w - kk[j].w);
    }
    lds_o[(qq * TT + lane) * NH + h] = ((a0 + a1) + (a2 + a3)) * nscale;
  }
  __syncthreads();

  // ---- coalesced nontemporal store: per q row, [t0..t0+31]x[h0..7] is
  // 1024 contiguous bytes in out[b,q,t,h] (h innermost, stride 1) ----
#pragma unroll
  for (int pass = 0; pass < 8; ++pass) {
    const int idx4 = pass * 256 + tid;   // float4 index into lds_o (2048 total)
    const int qq = idx4 >> 6;            // 64 float4 per q row (256 floats)
    const int w  = idx4 & 63;
    const f4 v = ((const f4*)lds_o)[idx4];
    const size_t goff = (((size_t)b * NS + q0 + qq) * NS + t0) * NH;  // floats
    __builtin_nontemporal_store(v, (f4*)(out + goff) + w);
  }
}

extern "C" void kernel_launch(void* const* d_in, const int* in_sizes, int n_in,
                              void* d_out, int out_size, void* d_ws, size_t ws_size,
                              hipStream_t stream) {
  const float* q = (const float*)d_in[0];
  const float* k = (const float*)d_in[1];
  float* out = (float*)d_out;
  (void)in_sizes; (void)n_in; (void)out_size; (void)d_ws; (void)ws_size;

  dim3 grid(NS / TT, NS / TQ, NB);   // (64, 64, 2)
  l1_scores_kernel<<<grid, 256, 0, stream>>>(q, k, out);
}